// DecoderRnn1_StandardLstmCell_layer_0_forward_35304631173361
// MI455X (gfx1250) — compile-verified
//
#include <hip/hip_runtime.h>

typedef float v2f __attribute__((ext_vector_type(2)));
typedef float v8f __attribute__((ext_vector_type(8)));

#define B_ROWS 1048576
#define IN_DIM 16
#define H_DIM  32
#define G_DIM  128   // 4*H

__device__ __forceinline__ float fast_sigmoid(float x) {
    return 1.0f / (1.0f + __expf(-x));
}
__device__ __forceinline__ float fast_tanh(float x) {
    // tanh(x) = 1 - 2/(exp(2x)+1)
    return 1.0f - 2.0f / (__expf(2.0f * x) + 1.0f);
}

// One wave handles a 16-row batch tile against the full 128-wide gate matrix.
// Weight B-fragments (pre-swizzled to WMMA lane layout) live in LDS.
__launch_bounds__(256)
__global__ void lstm_cell_wmma(const float* __restrict__ x,
                               const float* __restrict__ h,
                               const float* __restrict__ c,
                               const float* __restrict__ Wx,
                               const float* __restrict__ Wh,
                               const float* __restrict__ b,
                               float* __restrict__ out) {
    // 96 B-fragments (32 for Wx: nt*4+kg; 64 for Wh: nt*8+kg), 32 lanes each.
    __shared__ v2f wfrag[96 * 32];   // 24 KB

    const int tid = threadIdx.x;

    // Cooperative swizzle of weights into WMMA B-fragment layout:
    // lane ln: col n = ln&15, K rows = kg*4 + (ln>>4)*2 + {0,1}
    for (int e = tid; e < 96 * 32; e += 256) {
        const int frag = e >> 5;
        const int ln   = e & 31;
        const int kb   = (ln >> 4) << 1;
        const int n    = ln & 15;
        v2f w;
        if (frag < 32) {                 // Wx fragment
            const int nt = frag >> 2, kg = frag & 3;
            const int k = kg * 4 + kb, col = nt * 16 + n;
            w.x = Wx[k * G_DIM + col];
            w.y = Wx[(k + 1) * G_DIM + col];
        } else {                         // Wh fragment
            const int fh = frag - 32;
            const int nt = fh >> 3, kg = fh & 7;
            const int k = kg * 4 + kb, col = nt * 16 + n;
            w.x = Wh[k * G_DIM + col];
            w.y = Wh[(k + 1) * G_DIM + col];
        }
        wfrag[e] = w;
    }
    __syncthreads();

    const int lane = tid & 31;
    const int wave = tid >> 5;
    const int n    = lane & 15;
    const int hi   = lane >> 4;          // which half-wave
    const int kb   = hi << 1;

    const int tile    = blockIdx.x * 8 + wave;
    const int rowbase = tile * 16;

    // Bias per N-tile, one value per lane column (splat across accumulator rows).
    float bias[8];
    #pragma unroll
    for (int nt = 0; nt < 8; ++nt) bias[nt] = b[nt * 16 + n];

    // A-fragments: lane holds row (rowbase + n), K = kg*4 + kb + {0,1}
    const int arow = rowbase + n;
    v2f ax[4];
    #pragma unroll
    for (int kg = 0; kg < 4; ++kg)
        ax[kg] = *(const v2f*)(x + arow * IN_DIM + kg * 4 + kb);
    v2f ah[8];
    #pragma unroll
    for (int kg = 0; kg < 8; ++kg)
        ah[kg] = *(const v2f*)(h + arow * H_DIM + kg * 4 + kb);

    // gates = bias + x@Wx + h@Wh  via 96 v_wmma_f32_16x16x4_f32
    v8f acc[8];
    #pragma unroll
    for (int nt = 0; nt < 8; ++nt) {
        v8f a;
        #pragma unroll
        for (int r = 0; r < 8; ++r) a[r] = bias[nt];
        #pragma unroll
        for (int kg = 0; kg < 4; ++kg) {
            v2f bw = wfrag[(nt * 4 + kg) * 32 + lane];
            a = __builtin_amdgcn_wmma_f32_16x16x4_f32(
                    false, ax[kg], false, bw, (short)0, a, false, false);
        }
        #pragma unroll
        for (int kg = 0; kg < 8; ++kg) {
            v2f bw = wfrag[(32 + nt * 8 + kg) * 32 + lane];
            a = __builtin_amdgcn_wmma_f32_16x16x4_f32(
                    false, ah[kg], false, bw, (short)0, a, false, false);
        }
        acc[nt] = a;
    }

    // Recombine gates: tiles {0,1}=i, {2,3}=f, {4,5}=g, {6,7}=o.
    float* __restrict__ h_out = out;
    float* __restrict__ c_out = out + (size_t)B_ROWS * H_DIM;

    #pragma unroll
    for (int half = 0; half < 2; ++half) {
        const int col = half * 16 + n;
        v8f gi = acc[0 + half];
        v8f gf = acc[2 + half];
        v8f gg = acc[4 + half];
        v8f go = acc[6 + half];
        #pragma unroll
        for (int r = 0; r < 8; ++r) {
            const int row = rowbase + hi * 8 + r;   // C/D layout: VGPR r -> row r + 8*hi
            const float cv = c[row * H_DIM + col];
            const float iv = fast_sigmoid(gi[r]);
            const float fv = fast_sigmoid(gf[r]);
            const float gv = fast_tanh(gg[r]);
            const float ov = fast_sigmoid(go[r]);
            const float cn = iv * gv + fv * cv;
            const float hn = ov * fast_tanh(cn);
            h_out[row * H_DIM + col] = hn;
            c_out[row * H_DIM + col] = cn;
        }
    }
}

extern "C" void kernel_launch(void* const* d_in, const int* in_sizes, int n_in,
                              void* d_out, int out_size, void* d_ws, size_t ws_size,
                              hipStream_t stream) {
    const float* x  = (const float*)d_in[0];
    const float* h  = (const float*)d_in[1];
    const float* c  = (const float*)d_in[2];
    const float* Wx = (const float*)d_in[3];
    const float* Wh = (const float*)d_in[4];
    const float* b  = (const float*)d_in[5];
    float* out = (float*)d_out;

    dim3 grid(B_ROWS / 128);   // 8 waves/block * 16 rows/wave
    dim3 block(256);
    hipLaunchKernelGGL(lstm_cell_wmma, grid, block, 0, stream,
                       x, h, c, Wx, Wh, b, out);
}